// Points_6932077216522
// MI455X (gfx1250) — compile-verified
//
#include <hip/hip_runtime.h>
#include <math.h>

typedef __attribute__((ext_vector_type(2))) float v2f;
typedef __attribute__((ext_vector_type(4))) float v4f;
typedef __attribute__((ext_vector_type(8))) float v8f;

#define MT 2  // pixel (M) tiles per wave

// relu(1-d): d>=0 (PSD quadratic form) so pre_relu <= 1; clamp via single v_med3
#define RELU01(x) __builtin_amdgcn_fmed3f((x), 0.0f, 2.0f)

// One wave processes MT 16-pixel tiles of the canvas.
// GEMM1: features(16x8) @ coeffs(8x16 points) via V_WMMA_F32_16X16X4_F32 (x2/tile).
// GEMM2: P-reduction of relu(tile)*weighted_colors via packed FMAs, finished by
//        a reduce-scatter shfl_xor butterfly (live set halves every step).
__global__ void __launch_bounds__(256)
points_splat_kernel(const int* __restrict__ Hc, const int* __restrict__ Wc,
                    const float* __restrict__ loc,     // (P,2) [y,x]
                    const float* __restrict__ mo,      // (P,2,2)
                    const float* __restrict__ msfo,    // (P)
                    const float* __restrict__ colors,  // (P,3)
                    const float* __restrict__ alphas,  // (P)
                    float* __restrict__ out,           // (H,W,3)
                    int P, int P16, int npix) {
  extern __shared__ float smem[];
  v2f* s_ab  = (v2f*)smem;                 // 2*P16: (c0,c1) then (c2,c3)
  v2f* s_c45 = (v2f*)(smem + 4 * P16);     // P16:   (c4,c5)
  v4f* s_wc  = (v4f*)(smem + 6 * P16);     // P16:   (wc0,wc1,wc2,0)

  const int tid = threadIdx.x;
  const int H = Hc[0];
  const int W = Wc[0];

  // ---- per-point coefficient prep, zero-padded to P16 (block-cooperative) ----
  const float sqrtP = sqrtf((float)P);
  for (int p = tid; p < P16; p += blockDim.x) {
    if (p < P) {
      float ly = loc[2 * p + 0];
      float lx = loc[2 * p + 1];
      float m00 = mo[4 * p + 0], m01 = mo[4 * p + 1];
      float m10 = mo[4 * p + 2], m11 = mo[4 * p + 3];
      float s = 0.5f * sqrtP * __expf(msfo[p]);
      float T00 = m00 + s, T01 = m01, T10 = m10, T11 = m11 + s;
      float A = T00 * T00 + T01 * T01;
      float B = T10 * T10 + T11 * T11;
      float C = 2.0f * (T00 * T10 + T01 * T11);
      // pre_relu = c0*y^2 + c1*x^2 + c2*xy + c3*y + c4*x + c5  ( = 1 - d )
      s_ab[p]       = (v2f){-A, -B};
      s_ab[P16 + p] = (v2f){-C, 2.0f * A * ly + C * lx};
      s_c45[p]      = (v2f){2.0f * B * lx + C * ly,
                            1.0f - (A * ly * ly + B * lx * lx + C * ly * lx)};
      float al = alphas[p];
      s_wc[p] = (v4f){colors[3 * p + 0] * al, colors[3 * p + 1] * al,
                      colors[3 * p + 2] * al, 0.0f};
    } else {
      s_ab[p]       = (v2f){0.0f, 0.0f};
      s_ab[P16 + p] = (v2f){0.0f, 0.0f};
      s_c45[p]      = (v2f){0.0f, 0.0f};
      s_wc[p]       = (v4f){0.0f, 0.0f, 0.0f, 0.0f};
    }
  }
  __syncthreads();

  // ---- per-wave pixel tiles ----
  const int lane  = tid & 31;
  const bool hi   = lane >= 16;
  const int ln    = lane & 15;
  const int wv    = blockIdx.x * (blockDim.x >> 5) + (tid >> 5);
  const int tile0 = wv * MT;

  const float ratio = (float)W / (float)H;
  const float ysc = 2.0f / (float)(H - 1);
  const float xsc = 2.0f * ratio / (float)(W - 1);

  // A fragments for V_WMMA_F32_16X16X4_F32:
  //   lanes 0-15 : row M=lane,    VGPR0=K0, VGPR1=K1
  //   lanes16-31 : row M=lane-16, VGPR0=K2, VGPR1=K3
  v2f a1[MT], a2[MT];
#pragma unroll
  for (int mt = 0; mt < MT; ++mt) {
    int pix  = (tile0 + mt) * 16 + ln;
    int cpix = pix < npix ? pix : npix - 1;   // clamp, no divergence
    int h = cpix / W;
    int w = cpix - h * W;
    float y = -1.0f + ysc * (float)h;
    float x = -ratio + xsc * (float)w;
    a1[mt] = (v2f){hi ? x * y : y * y, hi ? y : x * x};  // K0=y^2|K2=xy, K1=x^2|K3=y
    a2[mt] = (v2f){hi ? 0.0f : x, hi ? 0.0f : 1.0f};     // K4=x |K6=0,  K5=1  |K7=0
  }

  float acc[MT][8][3];
#pragma unroll
  for (int mt = 0; mt < MT; ++mt)
#pragma unroll
    for (int r = 0; r < 8; ++r) {
      acc[mt][r][0] = 0.f; acc[mt][r][1] = 0.f; acc[mt][r][2] = 0.f;
    }

  const int abOff = hi ? P16 : 0;
  const int ntile = P16 >> 4;
#pragma unroll 2
  for (int t = 0; t < ntile; ++t) {
    int pt = t * 16 + ln;
    v2f b1  = s_ab[abOff + pt];   // (c0,c1) | (c2,c3)
    v2f c45 = s_c45[pt];          // unconditional load; select in VALU
    v4f wc  = s_wc[pt];
    v2f b2;
    b2.x = hi ? 0.0f : c45.x;     // K4 | K6=0
    b2.y = hi ? 0.0f : c45.y;     // K5 | K7=0

#pragma unroll
    for (int mt = 0; mt < MT; ++mt) {
      v8f c = {};
      c = __builtin_amdgcn_wmma_f32_16x16x4_f32(false, a1[mt], false, b1,
                                                (short)0, c, false, false);
      c = __builtin_amdgcn_wmma_f32_16x16x4_f32(false, a2[mt], false, b2,
                                                (short)0, c, false, false);
      // c[r]: lanes0-15 -> (M=r, N=lane); lanes16-31 -> (M=r+8, N=lane-16)
#pragma unroll
      for (int r = 0; r < 8; ++r) {
        float m = RELU01(c[r]);               // relu(1 - d), single v_med3
        acc[mt][r][0] = fmaf(m, wc.x, acc[mt][r][0]);
        acc[mt][r][1] = fmaf(m, wc.y, acc[mt][r][1]);
        acc[mt][r][2] = fmaf(m, wc.z, acc[mt][r][2]);
      }
    }
  }

  // ---- reduce-scatter over the 16 points per lane-group ----
  // Masks 8,4,2 partition on pixel-row bits; mask 1 partitions channels.
  // End state: lane ln owns pixel row (ln>>1); even ln: ch0(e0),ch1(e1); odd: ch2(e0).
  const bool s3 = (ln & 8) != 0;
  const bool s2 = (ln & 4) != 0;
  const bool s1 = (ln & 2) != 0;
  const bool s0 = (ln & 1) != 0;

#pragma unroll
  for (int mt = 0; mt < MT; ++mt) {
    // step mask 8: 24 live -> 12
    float b[4][3];
#pragma unroll
    for (int r = 0; r < 8; ++r)
#pragma unroll
      for (int ch = 0; ch < 3; ++ch)
        acc[mt][r][ch] += __shfl_xor(acc[mt][r][ch], 8, 32);
#pragma unroll
    for (int j = 0; j < 4; ++j)
#pragma unroll
      for (int ch = 0; ch < 3; ++ch)
        b[j][ch] = s3 ? acc[mt][j + 4][ch] : acc[mt][j][ch];

    // step mask 4: 12 -> 6
    float c2[2][3];
#pragma unroll
    for (int j = 0; j < 4; ++j)
#pragma unroll
      for (int ch = 0; ch < 3; ++ch)
        b[j][ch] += __shfl_xor(b[j][ch], 4, 32);
#pragma unroll
    for (int k = 0; k < 2; ++k)
#pragma unroll
      for (int ch = 0; ch < 3; ++ch)
        c2[k][ch] = s2 ? b[k + 2][ch] : b[k][ch];

    // step mask 2: 6 -> 3
    float d[3];
#pragma unroll
    for (int k = 0; k < 2; ++k)
#pragma unroll
      for (int ch = 0; ch < 3; ++ch)
        c2[k][ch] += __shfl_xor(c2[k][ch], 2, 32);
#pragma unroll
    for (int ch = 0; ch < 3; ++ch)
      d[ch] = s1 ? c2[1][ch] : c2[0][ch];

    // step mask 1: 3 -> 2 (even lanes: ch0,ch1 ; odd lanes: ch2)
#pragma unroll
    for (int ch = 0; ch < 3; ++ch)
      d[ch] += __shfl_xor(d[ch], 1, 32);
    float e0 = s0 ? d[2] : d[0];
    float e1 = d[1];

    // ---- store (all lanes active; row = ln>>1 within this half's 8 rows) ----
    int opix = (tile0 + mt) * 16 + (hi ? 8 : 0) + (ln >> 1);
    if (opix < npix) {
      float v0 = 1.0f / (1.0f + __expf(-4.0f * e0));
      if (!s0) {
        float v1 = 1.0f / (1.0f + __expf(-4.0f * e1));
        out[opix * 3 + 0] = v0;
        out[opix * 3 + 1] = v1;
      } else {
        out[opix * 3 + 2] = v0;
      }
    }
  }
}

extern "C" void kernel_launch(void* const* d_in, const int* in_sizes, int n_in,
                              void* d_out, int out_size, void* d_ws, size_t ws_size,
                              hipStream_t stream) {
  (void)n_in; (void)d_ws; (void)ws_size;
  const int*   Hc     = (const int*)d_in[0];
  const int*   Wc     = (const int*)d_in[1];
  const float* loc    = (const float*)d_in[2];
  const float* mo     = (const float*)d_in[3];
  const float* msfo   = (const float*)d_in[4];
  const float* colors = (const float*)d_in[5];
  const float* alphas = (const float*)d_in[6];
  float*       out    = (float*)d_out;

  const int P    = in_sizes[4];            // matrix_scale_factor_offsets has P elems
  const int P16  = (P + 15) & ~15;         // zero-padded point count
  const int npix = out_size / 3;           // H * W

  const int block = 256;                   // 8 waves
  const int pixPerBlock = (block / 32) * MT * 16;  // 256 pixels
  const int grid = (npix + pixPerBlock - 1) / pixPerBlock;
  const size_t smem = (size_t)(10 * P16) * sizeof(float);

  points_splat_kernel<<<grid, block, smem, stream>>>(Hc, Wc, loc, mo, msfo,
                                                     colors, alphas, out,
                                                     P, P16, npix);
}